// HumanFaceDetector_69922067579177
// MI455X (gfx1250) — compile-verified
//
#include <hip/hip_runtime.h>
#include <stdint.h>

#define N_BOXES_MAX 6000
#define IOU_THR 0.5f
#define CONF_THR 0.6f
#define NWMAX ((N_BOXES_MAX + 31) / 32)   // 188 mask words per row

// ---------------------------------------------------------------------------
// Kernel 1: stable descending rank (== argsort(-scores)) + scatter into
// sorted order. O(N^2) compares through an LDS tile: 36M ops, trivial.
// ---------------------------------------------------------------------------
__global__ void rank_scatter_kernel(const float* __restrict__ scores,
                                    const float4* __restrict__ boxes,
                                    float4* __restrict__ sboxes,
                                    float* __restrict__ sscores,
                                    int* __restrict__ order, int n) {
  __shared__ float tile[256];
  const int i = blockIdx.x * 256 + (int)threadIdx.x;
  const float si = (i < n) ? scores[i] : 0.0f;
  int rank = 0;
  for (int base = 0; base < n; base += 256) {
    const int j = base + (int)threadIdx.x;
    tile[threadIdx.x] = (j < n) ? scores[j] : 0.0f;
    __syncthreads();
    const int lim = min(256, n - base);
    for (int t = 0; t < lim; ++t) {
      const float sj = tile[t];
      const int jj = base + t;
      // stable descending: j precedes i if score greater, or equal with j < i
      rank += ((sj > si) || (sj == si && jj < i)) ? 1 : 0;
    }
    __syncthreads();
  }
  if (i < n) {
    sboxes[rank]  = boxes[i];
    sscores[rank] = si;
    order[rank]   = i;
  }
}

// ---------------------------------------------------------------------------
// Kernel 2: suppression bitmask.  One wave32 block per (32-row, 32-col) tile.
// Column boxes are staged into LDS with the CDNA5 async-to-LDS path
// (global_load_async_to_lds_b128 + s_wait_asynccnt); the block is a single
// wave so __syncthreads() lowers to S_NOP.
// mask[i*nw + cw] bit c == 1  iff  j = cw*32+c satisfies j>i, j<n, IoU>0.5.
// ---------------------------------------------------------------------------
__global__ void iou_mask_kernel(const float4* __restrict__ sboxes,
                                uint32_t* __restrict__ mask, int n, int nw) {
  __shared__ float4 colBoxes[32];
  const int lane = (int)threadIdx.x;        // 0..31, full wave active
  const int cw   = (int)blockIdx.x;         // column word
  const int i    = (int)blockIdx.y * 32 + lane;  // my row (sorted order)

  // --- async stage: each lane DMAs one column box (16B) into LDS ---
  {
    int col = cw * 32 + lane;
    if (col >= n) col = n - 1;              // clamp: stays in-bounds, bits masked later
    const uint64_t gaddr = (uint64_t)(uintptr_t)(const void*)(sboxes + col);
    const uint32_t laddr = (uint32_t)(uintptr_t)(const void*)(&colBoxes[lane]);
    asm volatile("global_load_async_to_lds_b128 %0, %1, off"
                 :: "v"(laddr), "v"(gaddr) : "memory");
  }
  // row box via the normal VMEM path (overlaps with the async transfer)
  const float4 rb = sboxes[min(i, n - 1)];
  asm volatile("s_wait_asynccnt 0" ::: "memory");
  __syncthreads();                           // single-wave: S_NOP

  const float ax1 = rb.x, ay1 = rb.y, ax2 = rb.z, ay2 = rb.w;
  const float areaA = (ax2 - ax1) * (ay2 - ay1);
  uint32_t bits = 0u;
#pragma unroll 8
  for (int c = 0; c < 32; ++c) {
    const float4 cb = colBoxes[c];           // LDS broadcast read
    const int j = cw * 32 + c;
    const float iw   = fminf(ax2, cb.z) - fmaxf(ax1, cb.x);
    const float ih   = fminf(ay2, cb.w) - fmaxf(ay1, cb.y);
    const float inter = fmaxf(iw, 0.0f) * fmaxf(ih, 0.0f);
    const float areaB = (cb.z - cb.x) * (cb.w - cb.y);
    const float iou   = inter / (areaA + areaB - inter);
    const bool sup = (j > i) && (j < n) && (iou > IOU_THR);
    bits |= (sup ? 1u : 0u) << c;
  }
  if (i < n) mask[(size_t)i * nw + cw] = bits;
}

// ---------------------------------------------------------------------------
// Kernel 3: the inherently serial greedy scan — now a SINGLE wave32.
// All lanes read the same `removed` word (LDS broadcast) so the keep
// decision is wave-uniform with no barrier; LDS ops within one wave are
// in-order (DScnt), so the OR->test dependence needs no barrier either.
// Row i+1 is prefetched (global_prefetch_b8) to hide the dependent-load
// latency of the serial chain.
// Tail: scatter keep-mask back to original indices with confidence filter.
// ---------------------------------------------------------------------------
__global__ void nms_scan_kernel(const uint32_t* __restrict__ mask,
                                const float* __restrict__ sscores,
                                const int* __restrict__ order,
                                float* __restrict__ out, int n, int nw) {
  __shared__ uint32_t removed[NWMAX];
  __shared__ uint32_t keepBits[NWMAX];
  const int lane = (int)threadIdx.x;        // 0..31, single wave
  for (int t = lane; t < nw; t += 32) {
    removed[t]  = 0u;
    keepBits[t] = 0u;
  }
  __syncthreads();                           // single-wave: S_NOP

  for (int i = 0; i < n; ++i) {
    // prefetch next mask row (752B = 12 cachelines) regardless of outcome
    if (i + 1 < n && lane < 12)
      __builtin_prefetch(mask + (size_t)(i + 1) * nw + (size_t)lane * 16, 0, 3);
    // wave-uniform keep decision: every lane reads the same LDS word
    const bool keep = ((removed[i >> 5] >> (i & 31)) & 1u) == 0u;
    if (keep) {
      if (lane == 0) keepBits[i >> 5] |= (1u << (i & 31));
      const uint32_t* row = mask + (size_t)i * nw;
      for (int t = lane; t < nw; t += 32)    // 6 coalesced words per lane
        removed[t] |= row[t];
    }
  }

  // scatter: order[] is a bijection over [0,n), so every output is written
  for (int r = lane; r < n; r += 32) {
    const bool k = (keepBits[r >> 5] >> (r & 31)) & 1u;
    const float s = sscores[r];
    out[order[r]] = (k && s >= CONF_THR) ? s : 0.0f;
  }
}

// ---------------------------------------------------------------------------
extern "C" void kernel_launch(void* const* d_in, const int* in_sizes, int n_in,
                              void* d_out, int out_size, void* d_ws, size_t ws_size,
                              hipStream_t stream) {
  const float* boxes  = (const float*)d_in[0];   // [N,4] f32
  const float* scores = (const float*)d_in[1];   // [N]   f32
  const int n  = in_sizes[1];                    // 6000
  const int nw = (n + 31) / 32;                  // 188

  // workspace carve-up (all 16B-aligned):
  //   sboxes  : n * 16B  (96000)   @ 0
  //   sscores : n * 4B   (24000)   @ 96KiB
  //   order   : n * 4B   (24000)   @ 128KiB
  //   mask    : n*nw*4B  (4.51MB)  @ 160KiB
  char* ws = (char*)d_ws;
  float4*   sboxes  = (float4*)(ws);
  float*    sscores = (float*) (ws + 96 * 1024);
  int*      order   = (int*)   (ws + 128 * 1024);
  uint32_t* mask    = (uint32_t*)(ws + 160 * 1024);

  rank_scatter_kernel<<<(n + 255) / 256, 256, 0, stream>>>(
      scores, (const float4*)boxes, sboxes, sscores, order, n);

  dim3 g2((unsigned)nw, (unsigned)((n + 31) / 32));
  iou_mask_kernel<<<g2, 32, 0, stream>>>(sboxes, mask, n, nw);

  nms_scan_kernel<<<1, 32, 0, stream>>>(mask, sscores, order, (float*)d_out, n, nw);
}